// MultiHeadAttention_81965155877627
// MI455X (gfx1250) — compile-verified
//
#include <hip/hip_runtime.h>
#include <stdint.h>

typedef __attribute__((ext_vector_type(2))) float v2f;
typedef __attribute__((ext_vector_type(8))) float v8f;

#define NHEAD 16

// ---- CDNA5 async global->LDS helpers (gfx1250, tracked by ASYNCcnt) --------
__device__ __forceinline__ void async_b128(unsigned lds_off, unsigned long long gaddr) {
  asm volatile("global_load_async_to_lds_b128 %0, %1, off"
               :: "v"(lds_off), "v"(gaddr) : "memory");
}
__device__ __forceinline__ void async_b32(unsigned lds_off, unsigned long long gaddr) {
  asm volatile("global_load_async_to_lds_b32 %0, %1, off"
               :: "v"(lds_off), "v"(gaddr) : "memory");
}
#define LDS_OFF(p) ((unsigned)(uintptr_t)(p))
#define GLB(p) ((unsigned long long)(uintptr_t)(p))

// ---------------------------------------------------------------------------
// Batched GEMM on the f32 WMMA pipe:  C = alpha * (A x B[T]) + bias
//   BT=true : B is row-major [N x K]  (Linear weights / K-matrix for QK^T)
//   BT=false: B is row-major [K x N]  (V-matrix for P x V)
// Tiles staged K-major ([row][k], PAD=4) so fragments are single ds_load_b64,
// staged via async global->LDS with double buffering (ASYNCcnt pipelining).
// ---------------------------------------------------------------------------
template <int BM, int BN, int BK, int WGM, int WGN, int WM, int WN, bool BT>
__global__ __launch_bounds__(256) void wmma_gemm(
    const float* __restrict__ A, int lda, long sAb, long sAh,
    const float* __restrict__ Bm, int ldb, long sBb, long sBh,
    const float* __restrict__ bias,
    float* __restrict__ C, int ldc, long sCb, long sCh,
    int M, int N, int Kdim, float alpha) {
  static_assert(WGM * WM * 16 == BM && WGN * WN * 16 == BN, "tile mismatch");
  constexpr int PAD = 4;

  const int bn0 = blockIdx.x * BN;
  const int bm0 = blockIdx.y * BM;
  const int z = blockIdx.z;
  const int zb = z / NHEAD, zh = z % NHEAD;
  A += (long)zb * sAb + (long)zh * sAh;
  Bm += (long)zb * sBb + (long)zh * sBh;
  C += (long)zb * sCb + (long)zh * sCh;

  __shared__ float Asl[2][BM][BK + PAD];  // K-major: [m][k]
  __shared__ float Bsl[2][BN][BK + PAD];  // K-major: [n][k]

  const int tid = threadIdx.x;
  const int wave = tid >> 5;
  const int lane = tid & 31;
  const int half = lane >> 4;  // which 16-lane half of the wave (K select)
  const int r = lane & 15;     // row (A) / col (B) / col (C) index in tile

  const int wm0 = (wave / WGN) * (WM * 16);
  const int wn0 = (wave % WGN) * (WN * 16);

  // per-thread async-op counts (all exact multiples of 256 threads here)
  constexpr int A_PT = (BM * (BK / 4)) / 256;                       // b128 ops
  constexpr int B_PT = BT ? (BN * (BK / 4)) / 256 : (BK * BN) / 256;
  constexpr int PEND = A_PT + B_PT;

  auto stage = [&](int buf, int k0) {
    #pragma unroll
    for (int t = 0; t < A_PT; ++t) {
      const int s = tid + t * 256;
      const int m = s / (BK / 4), kq = s % (BK / 4);
      async_b128(LDS_OFF(&Asl[buf][m][kq * 4]),
                 GLB(A + (long)(bm0 + m) * lda + k0 + kq * 4));
    }
    if constexpr (BT) {
      #pragma unroll
      for (int t = 0; t < B_PT; ++t) {
        const int s = tid + t * 256;
        const int n = s / (BK / 4), kq = s % (BK / 4);
        async_b128(LDS_OFF(&Bsl[buf][n][kq * 4]),
                   GLB(Bm + (long)(bn0 + n) * ldb + k0 + kq * 4));
      }
    } else {
      // transpose-scatter: global [k][n] -> LDS [n][k], per-lane LDS dest
      #pragma unroll
      for (int t = 0; t < B_PT; ++t) {
        const int s = tid + t * 256;
        const int k = s / BN, n = s % BN;
        async_b32(LDS_OFF(&Bsl[buf][n][k]),
                  GLB(Bm + (long)(k0 + k) * ldb + bn0 + n));
      }
    }
  };

  v8f acc[WM][WN] = {};

  const int nt = Kdim / BK;
  stage(0, 0);
  for (int t = 0; t < nt; ++t) {
    const int cur = t & 1;
    if (t + 1 < nt) {
      stage(cur ^ 1, (t + 1) * BK);
      asm volatile("s_wait_asynccnt %0" :: "n"(PEND) : "memory");
    } else {
      asm volatile("s_wait_asynccnt 0" ::: "memory");
    }
    __syncthreads();

    #pragma unroll
    for (int kk = 0; kk < BK; kk += 4) {
      v2f afrag[WM], bfrag[WN];
      // A frag: element j = A[r][kk + 2*half + j]  -> one ds_load_b64
      #pragma unroll
      for (int i = 0; i < WM; ++i)
        afrag[i] = *(const v2f*)&Asl[cur][wm0 + i * 16 + r][kk + 2 * half];
      // B frag: element j = B[kk + 2*half + j][r]  (K-major in LDS)
      #pragma unroll
      for (int j = 0; j < WN; ++j)
        bfrag[j] = *(const v2f*)&Bsl[cur][wn0 + j * 16 + r][kk + 2 * half];
      #pragma unroll
      for (int i = 0; i < WM; ++i)
        #pragma unroll
        for (int j = 0; j < WN; ++j)
          acc[i][j] = __builtin_amdgcn_wmma_f32_16x16x4_f32(
              false, afrag[i], false, bfrag[j], (short)0, acc[i][j], false, false);
    }
    __syncthreads();
  }

  // ---- epilogue: D row = e + 8*half, col = r (ISA 7.12.2 C/D layout) ----
  #pragma unroll
  for (int i = 0; i < WM; ++i) {
    #pragma unroll
    for (int j = 0; j < WN; ++j) {
      const int col = bn0 + wn0 + j * 16 + r;
      const float bv = bias ? bias[col] : 0.0f;
      #pragma unroll
      for (int e = 0; e < 8; ++e) {
        const int row = bm0 + wm0 + i * 16 + e + 8 * half;
        C[(long)row * ldc + col] = alpha * acc[i][j][e] + bv;
      }
    }
  }
}

// ---------------------------------------------------------------------------
// In-place row softmax over rows of length 2048. One 256-thread block per row,
// 8 elements per thread, wave32 shuffle reduce + 8-wave LDS reduce.
// ---------------------------------------------------------------------------
__global__ __launch_bounds__(256) void softmax_rows(float* __restrict__ attn) {
  const long row = blockIdx.x;
  float* p = attn + row * 2048L;
  const int tid = threadIdx.x;
  const int wave = tid >> 5;
  const int lane = tid & 31;

  float v[8];
  float m = -3.402823466e38f;
  #pragma unroll
  for (int i = 0; i < 8; ++i) {
    v[i] = p[tid + 256 * i];
    m = fmaxf(m, v[i]);
  }
  #pragma unroll
  for (int off = 16; off; off >>= 1) m = fmaxf(m, __shfl_xor(m, off, 32));

  __shared__ float smax[8];
  __shared__ float ssum[8];
  if (lane == 0) smax[wave] = m;
  __syncthreads();
  float bm = smax[0];
  #pragma unroll
  for (int i = 1; i < 8; ++i) bm = fmaxf(bm, smax[i]);

  float s = 0.0f;
  #pragma unroll
  for (int i = 0; i < 8; ++i) {
    v[i] = __expf(v[i] - bm);
    s += v[i];
  }
  #pragma unroll
  for (int off = 16; off; off >>= 1) s += __shfl_xor(s, off, 32);
  if (lane == 0) ssum[wave] = s;
  __syncthreads();
  float tot = 0.0f;
  #pragma unroll
  for (int i = 0; i < 8; ++i) tot += ssum[i];
  const float inv = 1.0f / tot;
  #pragma unroll
  for (int i = 0; i < 8; ++i) p[tid + 256 * i] = v[i] * inv;
}

// ---------------------------------------------------------------------------
extern "C" void kernel_launch(void* const* d_in, const int* in_sizes, int n_in,
                              void* d_out, int out_size, void* d_ws, size_t ws_size,
                              hipStream_t stream) {
  (void)in_sizes; (void)n_in; (void)out_size; (void)ws_size;
  constexpr long DM = 1024, Lq = 2048, TOK = 2 * 2048, BH = 32;

  const float* query = (const float*)d_in[0];
  const float* key   = (const float*)d_in[1];
  const float* value = (const float*)d_in[2];
  const float* Wq = (const float*)d_in[3];
  const float* bq = (const float*)d_in[4];
  const float* Wk = (const float*)d_in[5];
  const float* bk = (const float*)d_in[6];
  const float* Wv = (const float*)d_in[7];
  const float* bv = (const float*)d_in[8];
  const float* Wo = (const float*)d_in[9];
  const float* bo = (const float*)d_in[10];

  float* out  = (float*)d_out;                 // [B, L, 1024]
  float* attn = out + TOK * DM;                // [B, H, L, L]

  float* Qp = (float*)d_ws;                    // [B, L, 1024] 16 MB each
  float* Kp = Qp + TOK * DM;
  float* Vp = Kp + TOK * DM;
  float* Op = Vp + TOK * DM;                   // pre-Wo attention output

  dim3 block(256, 1, 1);

  // --- Q/K/V projections: [4096x1024] @ W^T ---
  {
    dim3 g(DM / 128, TOK / 128, 1);
    wmma_gemm<128, 128, 16, 2, 4, 4, 2, true><<<g, block, 0, stream>>>(
        query, (int)DM, 0L, 0L, Wq, (int)DM, 0L, 0L, bq,
        Qp, (int)DM, 0L, 0L, (int)TOK, (int)DM, (int)DM, 1.0f);
    wmma_gemm<128, 128, 16, 2, 4, 4, 2, true><<<g, block, 0, stream>>>(
        key, (int)DM, 0L, 0L, Wk, (int)DM, 0L, 0L, bk,
        Kp, (int)DM, 0L, 0L, (int)TOK, (int)DM, (int)DM, 1.0f);
    wmma_gemm<128, 128, 16, 2, 4, 4, 2, true><<<g, block, 0, stream>>>(
        value, (int)DM, 0L, 0L, Wv, (int)DM, 0L, 0L, bv,
        Vp, (int)DM, 0L, 0L, (int)TOK, (int)DM, (int)DM, 1.0f);
  }

  // --- scores = (1/8) * Q_h @ K_h^T, straight into attn region of d_out ---
  {
    dim3 g(Lq / 128, Lq / 128, BH);
    wmma_gemm<128, 128, 16, 2, 4, 4, 2, true><<<g, block, 0, stream>>>(
        Qp, (int)DM, Lq * DM, 64L, Kp, (int)DM, Lq * DM, 64L, nullptr,
        attn, (int)Lq, (long)NHEAD * Lq * Lq, Lq * Lq,
        (int)Lq, (int)Lq, 64, 0.125f);
  }

  // --- softmax in place over last dim ---
  softmax_rows<<<dim3((unsigned)(BH * Lq)), block, 0, stream>>>(attn);

  // --- P @ V_h -> Op laid out [B, L, H*dk] (head merge by layout) ---
  {
    dim3 g(64 / 64, Lq / 128, BH);
    wmma_gemm<128, 64, 32, 4, 2, 2, 2, false><<<g, block, 0, stream>>>(
        attn, (int)Lq, (long)NHEAD * Lq * Lq, Lq * Lq,
        Vp, (int)DM, Lq * DM, 64L, nullptr,
        Op, (int)DM, Lq * DM, 64L,
        (int)Lq, 64, (int)Lq, 1.0f);
  }

  // --- output projection ---
  {
    dim3 g(DM / 128, TOK / 128, 1);
    wmma_gemm<128, 128, 16, 2, 4, 4, 2, true><<<g, block, 0, stream>>>(
        Op, (int)DM, 0L, 0L, Wo, (int)DM, 0L, 0L, bo,
        out, (int)DM, 0L, 0L, (int)TOK, (int)DM, (int)DM, 1.0f);
  }
}